// Model_1580547969651
// MI455X (gfx1250) — compile-verified
//
#include <hip/hip_runtime.h>

// CDNA5 (gfx1250) wave32 WMMA types: 16x16x4 f32 -> A,B are 2 VGPRs/lane, C/D 8 VGPRs/lane.
typedef __attribute__((ext_vector_type(2))) float v2f;
typedef __attribute__((ext_vector_type(8))) float v8f;

#define BROWS 8192
#define FDIM  4096

// ---- kernel 1: yc_part[rc][f] = sum over 1024-row chunk rc of y[:,f], via V_WMMA_F32_16X16X4_F32
// A = all-ones 16x4  =>  D[i,j] = sum_k B[k,j]  (K-layout of B is irrelevant since A is constant)
// Two independent accumulators (rows {0..3} and {4..7} of each 8-row step) for matrix-pipe ILP.
#define YC_CHUNKS 8
#define YC_ROWS   (BROWS / YC_CHUNKS)   // 1024

__global__ void __launch_bounds__(256) colsum_wmma(const float* __restrict__ y,
                                                   float* __restrict__ yc_part) {
    const int lane = threadIdx.x & 31;
    const int wave = threadIdx.x >> 5;
    const int wg   = blockIdx.x * 8 + wave;      // 256 blocks * 8 waves = 2048 wave-tasks
    const int colTile  = wg & 255;               // 256 column tiles of 16
    const int rowChunk = wg >> 8;                // 8 row chunks
    const int half = lane >> 4;
    const int n    = lane & 15;
    const int col  = colTile * 16 + n;

    v2f ones; ones.x = 1.0f; ones.y = 1.0f;
    v8f c0 = {}, c1 = {};

    // B tile rows: v0 <- row (+2*half), v1 <- row (+2*half+1); lanes 0-15 coalesce along a row.
    const float* p = y + (long long)(rowChunk * YC_ROWS + 2 * half) * FDIM + col;
    for (int it = 0; it < YC_ROWS / 8; ++it) {
        const float* q = p + (long long)it * 8 * FDIM;
        v2f b0, b1;
        b0.x = q[0];
        b0.y = q[FDIM];
        b1.x = q[4 * FDIM];
        b1.y = q[5 * FDIM];
        c0 = __builtin_amdgcn_wmma_f32_16x16x4_f32(false, ones, false, b0,
                                                   (short)0, c0, false, false);
        c1 = __builtin_amdgcn_wmma_f32_16x16x4_f32(false, ones, false, b1,
                                                   (short)0, c1, false, false);
    }
    // Every row of D is identical; D[0, n] lives in element 0 on lanes 0-15.
    if (lane < 16) yc_part[rowChunk * FDIM + col] = c0[0] + c1[0];
}

// ---- kernel 2: yc[f] = sum_rc yc_part[rc][f]  (deterministic fixed-order reduce)
__global__ void reduce_yc(const float* __restrict__ yc_part, float* __restrict__ yc) {
    int f = blockIdx.x * blockDim.x + threadIdx.x;
    float s = 0.0f;
    #pragma unroll
    for (int i = 0; i < YC_CHUNKS; ++i) s += yc_part[i * FDIM + f];
    yc[f] = s;
}

// ---- kernel 3: s_part[kchunk][b] = x[b, kchunk] . yc[kchunk]  via WMMA
// A = x tile (documented 16x4 f32 layout), B = yc broadcast across all 16 columns.
// Two accumulators over even/odd k-quads.
#define KSPLIT  8
#define KCHUNK  (FDIM / KSPLIT)   // 512
#define TILE_K  64
#define RPB     128               // rows per block (8 waves * 16)
#define LDS_PITCH (TILE_K + 4)    // +4 floats keeps 16B alignment and kills bank conflicts

__global__ void __launch_bounds__(256) matvec_wmma(const float* __restrict__ x,
                                                   const float* __restrict__ yc,
                                                   float* __restrict__ s_part) {
    __shared__ float tile[RPB][LDS_PITCH];
    const int lane = threadIdx.x & 31;
    const int wave = threadIdx.x >> 5;
    const int half = lane >> 4;
    const int m    = lane & 15;
    const int rowbase = blockIdx.x * RPB;
    const int kbase0  = blockIdx.y * KCHUNK;

    v8f c0 = {}, c1 = {};
    for (int stg = 0; stg < KCHUNK / TILE_K; ++stg) {
        const int kb = kbase0 + stg * TILE_K;
        // Stage 128x64 fp32 tile: coalesced float4 loads, 16 float4 per row.
        #pragma unroll
        for (int i = 0; i < 8; ++i) {
            int idx = threadIdx.x + i * 256;   // float4 index
            int row = idx >> 4;
            int c4  = idx & 15;
            float4 v = *(const float4*)(x + (long long)(rowbase + row) * FDIM + kb + c4 * 4);
            *(float4*)&tile[row][c4 * 4] = v;
        }
        __syncthreads();

        const int r = wave * 16 + m;
        #pragma unroll
        for (int kk = 0; kk < TILE_K; kk += 8) {
            v2f a0, b0, a1, b1;
            // Documented A 16x4 layout: lanes<16: v0=K0, v1=K1; lanes>=16: v0=K2, v1=K3.
            a0.x = tile[r][kk + 2 * half];
            a0.y = tile[r][kk + 2 * half + 1];
            a1.x = tile[r][kk + 4 + 2 * half];
            a1.y = tile[r][kk + 4 + 2 * half + 1];
            // yc broadcast over N with matching K mapping.
            b0.x = yc[kb + kk + 2 * half];
            b0.y = yc[kb + kk + 2 * half + 1];
            b1.x = yc[kb + kk + 4 + 2 * half];
            b1.y = yc[kb + kk + 4 + 2 * half + 1];
            c0 = __builtin_amdgcn_wmma_f32_16x16x4_f32(false, a0, false, b0,
                                                       (short)0, c0, false, false);
            c1 = __builtin_amdgcn_wmma_f32_16x16x4_f32(false, a1, false, b1,
                                                       (short)0, c1, false, false);
        }
        __syncthreads();
    }
    // D columns are all identical; VGPR v holds M=v (lanes 0-15) / M=v+8 (lanes 16-31).
    if ((lane & 15) == 0) {
        int mbase = rowbase + wave * 16 + half * 8;
        float* dst = s_part + (long long)blockIdx.y * BROWS + mbase;
        #pragma unroll
        for (int v = 0; v < 8; ++v) dst[v] = c0[v] + c1[v];
    }
}

// ---- kernel 4: out[b] = clip(hardswish(sum_k s_part[k][b]) + noise[b], -0.5, 0.5)
__global__ void epilogue(const float* __restrict__ s_part, const float* __restrict__ noise,
                         float* __restrict__ out) {
    int b = blockIdx.x * blockDim.x + threadIdx.x;
    float s = 0.0f;
    #pragma unroll
    for (int i = 0; i < KSPLIT; ++i) s += s_part[(long long)i * BROWS + b];
    float t = fminf(fmaxf(s + 3.0f, 0.0f), 6.0f);
    float h = s * t * (1.0f / 6.0f);           // hardswish; logsumexp on singleton = identity
    float r = h + noise[b];
    out[b] = fminf(fmaxf(r, -0.5f), 0.5f);     // hardtanh(-0.5, 0.5)
}

extern "C" void kernel_launch(void* const* d_in, const int* in_sizes, int n_in,
                              void* d_out, int out_size, void* d_ws, size_t ws_size,
                              hipStream_t stream) {
    const float* x     = (const float*)d_in[0];   // (8192, 4096) fp32
    const float* y     = (const float*)d_in[1];   // (8192, 4096) fp32
    const float* noise = (const float*)d_in[2];   // (8192, 1) fp32
    float* out = (float*)d_out;                   // (8192, 1) fp32

    float* ws      = (float*)d_ws;
    float* yc_part = ws;                          // 8 * 4096 floats
    float* yc      = ws + YC_CHUNKS * FDIM;       // 4096 floats
    float* s_part  = yc + FDIM;                   // 8 * 8192 floats (~400 KB total ws)

    colsum_wmma<<<256, 256, 0, stream>>>(y, yc_part);
    reduce_yc<<<FDIM / 256, 256, 0, stream>>>(yc_part, yc);
    matvec_wmma<<<dim3(BROWS / RPB, KSPLIT), 256, 0, stream>>>(x, yc, s_part);
    epilogue<<<BROWS / 256, 256, 0, stream>>>(s_part, noise, out);
}